// StructureNet_74002286510097
// MI455X (gfx1250) — compile-verified
//
#include <hip/hip_runtime.h>
#include <math.h>

// ---------------------------------------------------------------------------
// StructureNet (IPA) for gfx1250 — fp32 WMMA (V_WMMA_F32_16X16X4_F32) path.
// LDS-staged shared A operands (b128 fills, conflict-free padded strides),
// dual-pointer strided B walks (no per-iteration 64-bit multiplies),
// no per-load masking (clamped addresses; garbage tiles never stored).
// B=1, N=512, CS=384, CZ=128, L=4, H=12, C=16, PQ=4, PV=8
// ---------------------------------------------------------------------------

static constexpr int kN   = 512;
static constexpr int kCS  = 384;
static constexpr int kCZ  = 128;
static constexpr int kH   = 12;
static constexpr int kC   = 16;
static constexpr int kPQ  = 4;
static constexpr int kPV  = 8;
static constexpr int kL   = 4;
static constexpr int kNN  = kN * kN;
static constexpr int kCAT = kH * kC + 4 * kH * kPV + kH * kCZ; // 2112
static constexpr int kQD  = kH * kC;          // 192
static constexpr int kKVD = 2 * kH * kC;      // 384
static constexpr int kQPD = kH * kPQ * 3;     // 144
static constexpr int kKPD = kH * (kPQ + kPV) * 3; // 432

typedef float v2f __attribute__((ext_vector_type(2)));
typedef float v8f __attribute__((ext_vector_type(8)));

#define WMMA_F32(acc, a, b) \
  __builtin_amdgcn_wmma_f32_16x16x4_f32(false, (a), false, (b), (short)0, (acc), false, false)

// ---------------------------------------------------------------------------
// Generic WMMA GEMM:  C = A(MxK) @ W(KxNc) [+bias] [+resid] [relu]
// One wave per 16x16 C tile; 4 waves/block share an LDS-staged A chunk
// (16 rows x 64 K, stride 68 floats -> bank = 4*m + k, conflict-free).
// K must be a multiple of 64 here (384 or 2112 in this model).
// ---------------------------------------------------------------------------
__global__ void gemm_wmma(const float* __restrict__ A, int lda,
                          const float* __restrict__ W, int ldw,
                          const float* __restrict__ bias,
                          const float* __restrict__ resid,
                          float* __restrict__ Cout, int ldc,
                          int K, int Nc, int relu)
{
  __shared__ float At[16][68];
  const int tid  = threadIdx.x;
  const int wid  = tid >> 5;
  const int lane = tid & 31;
  const int nt   = blockIdx.x * 4 + wid;
  const int n0   = nt * 16;
  const int m0   = blockIdx.y * 16;
  const int m    = lane & 15;
  const int kh   = lane >> 4;
  const int col  = n0 + m;
  const bool colOK = (col < Nc);
  const int colc = colOK ? col : 0;          // clamped address; garbage cols unused

  // Staging coordinates (b128): thread loads rows rr and rr+8, 4 floats each.
  const int srr = tid >> 4;                  // 0..7
  const int sc4 = (tid & 15) << 2;           // 0..60

  // B operand: dual pointers, advanced by 4*ldw per K-step.
  const float* wp0 = W + (size_t)(2 * kh) * ldw + colc;
  const float* wp1 = wp0 + ldw;
  const size_t wstep = (size_t)4 * ldw;

  v8f acc = {0.f,0.f,0.f,0.f,0.f,0.f,0.f,0.f};
  for (int kc = 0; kc < K; kc += 64) {
    __syncthreads();
    *(float4*)&At[srr][sc4] =
        *(const float4*)&A[(size_t)(m0 + srr) * lda + kc + sc4];
    *(float4*)&At[srr + 8][sc4] =
        *(const float4*)&A[(size_t)(m0 + srr + 8) * lda + kc + sc4];
    __syncthreads();
#pragma unroll
    for (int kk = 0; kk < 64; kk += 4) {
      const int ka = kk + 2 * kh;
      v2f a, b;
      a.x = At[m][ka];
      a.y = At[m][ka + 1];
      b.x = wp0[0];
      b.y = wp1[0];
      wp0 += wstep; wp1 += wstep;
      acc = WMMA_F32(acc, a, b);
    }
  }
  if (colOK) {
    const float bv = bias ? bias[col] : 0.f;
#pragma unroll
    for (int r = 0; r < 8; ++r) {
      const int row = m0 + r + 8 * kh;
      float v = acc[r] + bv;
      if (resid) v += resid[(size_t)row * ldc + col];
      if (relu)  v = fmaxf(v, 0.f);
      Cout[(size_t)row * ldc + col] = v;
    }
  }
}

// ---------------------------------------------------------------------------
// Pair LayerNorm stats: mean / rstd per (i,j) row of CZ=128. One wave per row,
// one b128 load per lane.
// ---------------------------------------------------------------------------
__global__ void pair_stats(const float* __restrict__ pair,
                           float* __restrict__ mu, float* __restrict__ rs)
{
  const int wid = threadIdx.x >> 5, lane = threadIdx.x & 31;
  const size_t p = (size_t)blockIdx.x * (blockDim.x >> 5) + wid;
  const float4 v = *(const float4*)&pair[p * kCZ + lane * 4];
  float s  = v.x + v.y + v.z + v.w;
  float sq = v.x * v.x + v.y * v.y + v.z * v.z + v.w * v.w;
#pragma unroll
  for (int o = 16; o > 0; o >>= 1) {
    s  += __shfl_xor(s,  o, 32);
    sq += __shfl_xor(sq, o, 32);
  }
  if (lane == 0) {
    const float m = s * (1.f / kCZ);
    mu[p] = m;
    rs[p] = rsqrtf(sq * (1.f / kCZ) - m * m + 1e-5f);
  }
}

// ---------------------------------------------------------------------------
// Row LayerNorm over last dim D (in-place safe). One block per row.
// ---------------------------------------------------------------------------
__global__ void ln_rows(const float* __restrict__ in, float* __restrict__ out,
                        const float* __restrict__ g, const float* __restrict__ b,
                        int D)
{
  const int row = blockIdx.x;
  const float* x = in + (size_t)row * D;
  float* o = out + (size_t)row * D;
  __shared__ float rsum[256], rsq[256];
  const int tid = threadIdx.x;
  float s = 0.f, sq = 0.f;
  for (int c = tid; c < D; c += blockDim.x) { float v = x[c]; s += v; sq += v * v; }
  rsum[tid] = s; rsq[tid] = sq;
  __syncthreads();
  for (int st = blockDim.x >> 1; st > 0; st >>= 1) {
    if (tid < st) { rsum[tid] += rsum[tid + st]; rsq[tid] += rsq[tid + st]; }
    __syncthreads();
  }
  const float mean = rsum[0] / D;
  const float var  = rsq[0] / D - mean * mean;
  const float rstd = rsqrtf(var + 1e-5f);
  for (int c = tid; c < D; c += blockDim.x) {
    float v = x[c];
    o[c] = (v - mean) * rstd * g[c] + b[c];
  }
}

// ---------------------------------------------------------------------------
// bmat(i,j,h) = LN(z[i,j,:]) . wb[:,h] + bpb[h]   — WMMA with LN fused in A.
// pair rows are used exactly once -> stream straight from L2, no staging.
// ---------------------------------------------------------------------------
__global__ void zb_gemm(const float* __restrict__ pair,
                        const float* __restrict__ mu, const float* __restrict__ rs,
                        const float* __restrict__ g,  const float* __restrict__ be,
                        const float* __restrict__ wb, const float* __restrict__ bpb,
                        float* __restrict__ bmat)
{
  const int wid = threadIdx.x >> 5, lane = threadIdx.x & 31;
  const int mt = blockIdx.x * (blockDim.x >> 5) + wid;
  const int m0 = mt * 16;
  const int m = lane & 15, kh = lane >> 4;
  const size_t r = (size_t)m0 + m;
  const float muv = mu[r], rsv = rs[r];
  const float* ap  = pair + r * kCZ + 2 * kh;
  const float* gp  = g  + 2 * kh;
  const float* bep = be + 2 * kh;
  const int mc = (m < kH) ? m : (kH - 1);    // clamped; garbage cols unused
  const float* wp = wb + 2 * kh * kH + mc;   // walks by 4*kH = 48 floats
  v8f acc = {0.f,0.f,0.f,0.f,0.f,0.f,0.f,0.f};
#pragma unroll 4
  for (int k0 = 0; k0 < kCZ; k0 += 4) {
    v2f a, b;
    a.x = (ap[0] - muv) * rsv * gp[0] + bep[0];
    a.y = (ap[1] - muv) * rsv * gp[1] + bep[1];
    b.x = wp[0];
    b.y = wp[kH];
    ap += 4; gp += 4; bep += 4; wp += 4 * kH;
    acc = WMMA_F32(acc, a, b);
  }
  if (m < kH) {
    const float bb = bpb[m];
#pragma unroll
    for (int rr = 0; rr < 8; ++rr) {
      const size_t row = (size_t)m0 + rr + 8 * kh;
      bmat[row * kH + m] = acc[rr] + bb;
    }
  }
}

// ---------------------------------------------------------------------------
// Point frame transform: qpg/kvpg = R @ p + 0.1*t. One block per residue.
// ---------------------------------------------------------------------------
__global__ void point_transform(const float* __restrict__ qp,
                                const float* __restrict__ kvp,
                                const float* __restrict__ Rb,
                                const float* __restrict__ tb,
                                float* __restrict__ qpg, float* __restrict__ kvpg)
{
  const int n = blockIdx.x;
  __shared__ float Rs[9], ts[3];
  if (threadIdx.x < 9) Rs[threadIdx.x] = Rb[n * 9 + threadIdx.x];
  if (threadIdx.x < 3) ts[threadIdx.x] = tb[n * 3 + threadIdx.x] * 0.1f;
  __syncthreads();
  const int idx = threadIdx.x;
  if (idx < kH * (kPQ + kPV)) {
    const float* s = kvp + (size_t)n * kKPD + idx * 3;
    float x = s[0], y = s[1], z = s[2];
    float* d = kvpg + (size_t)n * kKPD + idx * 3;
    d[0] = Rs[0]*x + Rs[1]*y + Rs[2]*z + ts[0];
    d[1] = Rs[3]*x + Rs[4]*y + Rs[5]*z + ts[1];
    d[2] = Rs[6]*x + Rs[7]*y + Rs[8]*z + ts[2];
  }
  if (idx < kH * kPQ) {
    const float* s = qp + (size_t)n * kQPD + idx * 3;
    float x = s[0], y = s[1], z = s[2];
    float* d = qpg + (size_t)n * kQPD + idx * 3;
    d[0] = Rs[0]*x + Rs[1]*y + Rs[2]*z + ts[0];
    d[1] = Rs[3]*x + Rs[4]*y + Rs[5]*z + ts[1];
    d[2] = Rs[6]*x + Rs[7]*y + Rs[8]*z + ts[2];
  }
}

// Sum-of-squares of query / key points per (n,h) — for |q-k|^2 expansion.
__global__ void point_sums(const float* __restrict__ qpg,
                           const float* __restrict__ kvpg,
                           float* __restrict__ sqq, float* __restrict__ sqk)
{
  const int t = blockIdx.x * blockDim.x + threadIdx.x;
  if (t >= kN * kH) return;
  const int n = t / kH, h = t % kH;
  float s = 0.f;
  const float* q = qpg + (size_t)n * kQPD + h * 12;
#pragma unroll
  for (int k = 0; k < 12; ++k) s += q[k] * q[k];
  sqq[t] = s;
  s = 0.f;
  const float* k2 = kvpg + (size_t)n * kKPD + h * 36;
#pragma unroll
  for (int k = 0; k < 12; ++k) s += k2[k] * k2[k];
  sqk[t] = s;
}

// ---------------------------------------------------------------------------
// Attention logits: q.k (WMMA K=16), point dot (WMMA K=12), + pair bias,
// + point distance term, + mask. One wave per (h, 16x16 tile of (i,j)).
// ---------------------------------------------------------------------------
__global__ void attn_logits(const float* __restrict__ q,
                            const float* __restrict__ kv,
                            const float* __restrict__ qpg,
                            const float* __restrict__ kvpg,
                            const float* __restrict__ sqq,
                            const float* __restrict__ sqk,
                            const float* __restrict__ bmat,
                            const float* __restrict__ maskp,
                            const float* __restrict__ head_w,
                            float* __restrict__ abuf)
{
  const int h  = blockIdx.z;
  const int i0 = blockIdx.y * 16, j0 = blockIdx.x * 16;
  const int lane = threadIdx.x & 31;
  const int m = lane & 15, kh = lane >> 4;

  const float* qrow = q  + (size_t)(i0 + m) * kQD  + h * kC;
  const float* krow = kv + (size_t)(j0 + m) * kKVD + h * 2 * kC;  // k half
  v8f aqk = {0.f,0.f,0.f,0.f,0.f,0.f,0.f,0.f};
#pragma unroll
  for (int k0 = 0; k0 < kC; k0 += 4) {
    const int ka = k0 + 2 * kh;
    v2f a, b;
    a.x = qrow[ka]; a.y = qrow[ka + 1];
    b.x = krow[ka]; b.y = krow[ka + 1];
    aqk = WMMA_F32(aqk, a, b);
  }
  const float* qprow = qpg  + (size_t)(i0 + m) * kQPD + h * 12;
  const float* kprow = kvpg + (size_t)(j0 + m) * kKPD + h * 36;
  v8f apt = {0.f,0.f,0.f,0.f,0.f,0.f,0.f,0.f};
#pragma unroll
  for (int k0 = 0; k0 < 12; k0 += 4) {
    const int ka = k0 + 2 * kh;
    v2f a, b;
    a.x = qprow[ka]; a.y = qprow[ka + 1];
    b.x = kprow[ka]; b.y = kprow[ka + 1];
    apt = WMMA_F32(apt, a, b);
  }
  const float s1 = 0.14433756729740643f;   // 1/sqrt(3*C)
  const float c1 = 0.57735026918962576f;   // sqrt(1/3)
  const float wC = 0.13608276348795434f;   // sqrt(1/(3*PQ*9/2))
  const float gamma = log1pf(expf(head_w[h])) * wC;
  const int j = j0 + m;
  const float mj = maskp[j];
  const float sqkj = sqk[j * kH + h];
#pragma unroll
  for (int r = 0; r < 8; ++r) {
    const int i = i0 + r + 8 * kh;
    float val = aqk[r] * s1 + c1 * bmat[((size_t)i * kN + j) * kH + h];
    const float d2 = sqq[i * kH + h] + sqkj - 2.f * apt[r];
    val += -0.5f * d2 * gamma;
    val += (maskp[i] * mj - 1.f) * 1e9f;
    abuf[(size_t)h * kNN + (size_t)i * kN + j] = val;
  }
}

// Softmax over j (512) for each (h,i) row.
__global__ void softmax_rows(float* __restrict__ abuf)
{
  float* p = abuf + (size_t)blockIdx.x * kN;
  __shared__ float red[256];
  const int tid = threadIdx.x;
  const float x0 = p[tid], x1 = p[tid + 256];
  red[tid] = fmaxf(x0, x1);
  __syncthreads();
  for (int s = 128; s > 0; s >>= 1) {
    if (tid < s) red[tid] = fmaxf(red[tid], red[tid + s]);
    __syncthreads();
  }
  const float mx = red[0];
  __syncthreads();
  const float e0 = expf(x0 - mx), e1 = expf(x1 - mx);
  red[tid] = e0 + e1;
  __syncthreads();
  for (int s = 128; s > 0; s >>= 1) {
    if (tid < s) red[tid] += red[tid + s];
    __syncthreads();
  }
  const float inv = 1.f / red[0];
  p[tid] = e0 * inv;
  p[tid + 256] = e1 * inv;
}

// ---------------------------------------------------------------------------
// o = a @ v  and  opt = a @ vpg  fused: per (i-tile, head), 3 waves cover the
// 40 output cols (16 + 24). A tile (16 x 512) staged once in LDS (b128),
// shared by the 3 waves (stride 516 -> conflict-free b64 reads).
// ---------------------------------------------------------------------------
__global__ void av_gemm(const float* __restrict__ abuf,
                        const float* __restrict__ kv,
                        const float* __restrict__ kvpg,
                        float* __restrict__ cat, float* __restrict__ optbuf)
{
  __shared__ float Avs[16][516];
  const int h  = blockIdx.y;
  const int m0 = blockIdx.x * 16;
  const int tid = threadIdx.x;
  const int wid = tid >> 5, lane = tid & 31;
  const int m = lane & 15, kh = lane >> 4;

  for (int idx4 = tid; idx4 < 16 * (kN / 4); idx4 += 96) {
    const int rr = idx4 >> 7;
    const int c4 = (idx4 & 127) << 2;
    *(float4*)&Avs[rr][c4] =
        *(const float4*)&abuf[(size_t)h * kNN + (size_t)(m0 + rr) * kN + c4];
  }
  __syncthreads();

  const int n0 = wid * 16;
  const int colL = n0 + m;                 // 0..47, valid < 40
  const int cc = colL < 40 ? colL : 39;    // clamped; garbage cols unused
  const float* bbase;
  int bstride;
  if (cc < 16) { bbase = kv   + (h * 2 + 1) * kC + cc;       bstride = kKVD; }
  else         { bbase = kvpg + h * 36 + 12 + (cc - 16);     bstride = kKPD; }
  const float* bp0 = bbase + (size_t)(2 * kh) * bstride;
  const float* bp1 = bp0 + bstride;
  const size_t bstep = (size_t)4 * bstride;

  v8f acc = {0.f,0.f,0.f,0.f,0.f,0.f,0.f,0.f};
#pragma unroll 4
  for (int k0 = 0; k0 < kN; k0 += 4) {
    const int ka = k0 + 2 * kh;
    v2f a, b;
    a.x = Avs[m][ka]; a.y = Avs[m][ka + 1];
    b.x = bp0[0];
    b.y = bp1[0];
    bp0 += bstep; bp1 += bstep;
    acc = WMMA_F32(acc, a, b);
  }
#pragma unroll
  for (int r = 0; r < 8; ++r) {
    const int row = m0 + r + 8 * kh;
    if (colL < 16)
      cat[(size_t)row * kCAT + h * kC + colL] = acc[r];
    else if (colL < 40)
      optbuf[(size_t)row * (kH * 24) + h * 24 + (colL - 16)] = acc[r];
  }
}

// ---------------------------------------------------------------------------
// opair[i,h,c] = sum_j a[h,i,j] * LN(z[i,j,c]) — per-i GEMM (12x128, K=512).
// One 8-wave block per residue i; the 12x512 attention slice staged once in
// LDS (b128) and shared by all 8 column-tile waves.
// ---------------------------------------------------------------------------
__global__ void opair_gemm(const float* __restrict__ abuf,
                           const float* __restrict__ pair,
                           const float* __restrict__ mu,
                           const float* __restrict__ rs,
                           const float* __restrict__ g,
                           const float* __restrict__ be,
                           float* __restrict__ cat)
{
  __shared__ float Als[12][516];
  const int i = blockIdx.x;
  const int tid = threadIdx.x;
  const int wid = tid >> 5, lane = tid & 31;
  const int m = lane & 15, kh = lane >> 4;

  for (int idx4 = tid; idx4 < 12 * (kN / 4); idx4 += 256) {
    const int rr = idx4 >> 7;
    const int c4 = (idx4 & 127) << 2;
    *(float4*)&Als[rr][c4] =
        *(const float4*)&abuf[(size_t)rr * kNN + (size_t)i * kN + c4];
  }
  __syncthreads();

  const int n0 = wid * 16;
  const int c = n0 + m;                 // < 128 always
  const float gv = g[c], bv = be[c];
  const int mr = (m < kH) ? m : 0;      // clamped; garbage rows unused
  const float* bp0 = pair + (size_t)i * kN * kCZ + c + (size_t)(2 * kh) * kCZ;
  const float* bp1 = bp0 + kCZ;
  const float* mup = mu + (size_t)i * kN + 2 * kh;
  const float* rsp = rs + (size_t)i * kN + 2 * kh;
  v8f acc = {0.f,0.f,0.f,0.f,0.f,0.f,0.f,0.f};
#pragma unroll 4
  for (int k0 = 0; k0 < kN; k0 += 4) {
    const int ka = k0 + 2 * kh;
    v2f a, b;
    a.x = Als[mr][ka];
    a.y = Als[mr][ka + 1];
    b.x = (bp0[0] - mup[0]) * rsp[0] * gv + bv;
    b.y = (bp1[0] - mup[1]) * rsp[1] * gv + bv;
    bp0 += 4 * kCZ; bp1 += 4 * kCZ; mup += 4; rsp += 4;
    acc = WMMA_F32(acc, a, b);
  }
#pragma unroll
  for (int r = 0; r < 8; ++r) {
    const int hh = r + 8 * kh;
    if (hh < kH)
      cat[(size_t)i * kCAT + 576 + hh * kCZ + c] = acc[r];
  }
}

// ---------------------------------------------------------------------------
// Inverse-frame transform of opt, writes optf (transposed) + onorm into cat.
// ---------------------------------------------------------------------------
__global__ void opt_post(const float* __restrict__ optbuf,
                         const float* __restrict__ Rb,
                         const float* __restrict__ tb,
                         float* __restrict__ cat)
{
  const int t = blockIdx.x * blockDim.x + threadIdx.x;
  if (t >= kN * kH * kPV) return;
  const int n = t / (kH * kPV);
  const int hp = t % (kH * kPV);
  const float* R = Rb + n * 9;
  const float tx = tb[n * 3 + 0] * 0.1f;
  const float ty = tb[n * 3 + 1] * 0.1f;
  const float tz = tb[n * 3 + 2] * 0.1f;
  const float* o = optbuf + (size_t)n * (kH * 24) + hp * 3;
  const float vx = o[0] - tx, vy = o[1] - ty, vz = o[2] - tz;
  const float lx = R[0]*vx + R[3]*vy + R[6]*vz;
  const float ly = R[1]*vx + R[4]*vy + R[7]*vz;
  const float lz = R[2]*vx + R[5]*vy + R[8]*vz;
  const float nr = sqrtf(lx*lx + ly*ly + lz*lz + 1e-8f);
  float* cr = cat + (size_t)n * kCAT;
  cr[192 + 0 * 96 + hp] = lx;
  cr[192 + 1 * 96 + hp] = ly;
  cr[192 + 2 * 96 + hp] = lz;
  cr[480 + hp] = nr;
}

// Quaternion backbone update: R,t <- composed frames.
__global__ void frame_update(float* __restrict__ RB, float* __restrict__ TB,
                             const float* __restrict__ UPD)
{
  const int n = blockIdx.x * blockDim.x + threadIdx.x;
  if (n >= kN) return;
  const float* u = UPD + n * 6;
  float qb = u[0], qc = u[1], qd = u[2];
  const float tx = u[3] * 10.f, ty = u[4] * 10.f, tz = u[5] * 10.f;
  const float inv = rsqrtf(1.f + qb*qb + qc*qc + qd*qd);
  const float qa = inv;
  qb *= inv; qc *= inv; qd *= inv;
  const float Ru[9] = {
    qa*qa + qb*qb - qc*qc - qd*qd, 2.f*(qb*qc - qa*qd),           2.f*(qb*qd + qa*qc),
    2.f*(qb*qc + qa*qd),           qa*qa - qb*qb + qc*qc - qd*qd, 2.f*(qc*qd - qa*qb),
    2.f*(qb*qd - qa*qc),           2.f*(qc*qd + qa*qb),           qa*qa - qb*qb - qc*qc + qd*qd };
  float* Rp = RB + n * 9;
  float* tp = TB + n * 3;
  float R0[9];
#pragma unroll
  for (int i = 0; i < 9; ++i) R0[i] = Rp[i];
  const float t0[3] = {tp[0], tp[1], tp[2]};
#pragma unroll
  for (int i = 0; i < 3; ++i)
    tp[i] = R0[i*3+0]*tx + R0[i*3+1]*ty + R0[i*3+2]*tz + t0[i];
#pragma unroll
  for (int i = 0; i < 3; ++i)
#pragma unroll
    for (int j = 0; j < 3; ++j)
      Rp[i*3+j] = R0[i*3+0]*Ru[0*3+j] + R0[i*3+1]*Ru[1*3+j] + R0[i*3+2]*Ru[2*3+j];
}

__global__ void copy_f32(float* __restrict__ dst, const float* __restrict__ src, int n)
{
  const int i = blockIdx.x * blockDim.x + threadIdx.x;
  if (i < n) dst[i] = src[i];
}

// ---------------------------------------------------------------------------
// Host-side orchestration
// ---------------------------------------------------------------------------
extern "C" void kernel_launch(void* const* d_in, const int* in_sizes, int n_in,
                              void* d_out, int out_size, void* d_ws, size_t ws_size,
                              hipStream_t stream)
{
  (void)in_sizes; (void)n_in; (void)out_size; (void)ws_size;
  const float* single_rep = (const float*)d_in[0];
  const float* pair_rep   = (const float*)d_in[1];
  const float* rot        = (const float*)d_in[2];
  const float* trans      = (const float*)d_in[3];
  const float* maskp      = (const float*)d_in[4];
  const float* sr_ln_g    = (const float*)d_in[5];
  const float* sr_ln_b    = (const float*)d_in[6];
  const float* pr_ln_g    = (const float*)d_in[7];
  const float* pr_ln_b    = (const float*)d_in[8];
  const float* sr_w       = (const float*)d_in[9];
  const float* sr_b       = (const float*)d_in[10];
  const float* wq         = (const float*)d_in[11];
  const float* bq         = (const float*)d_in[12];
  const float* wkv        = (const float*)d_in[13];
  const float* bkv        = (const float*)d_in[14];
  const float* wqp        = (const float*)d_in[15];
  const float* bqp        = (const float*)d_in[16];
  const float* wkvp       = (const float*)d_in[17];
  const float* bkvp       = (const float*)d_in[18];
  const float* wb         = (const float*)d_in[19];
  const float* bpb        = (const float*)d_in[20];
  const float* head_w     = (const float*)d_in[21];
  const float* wo         = (const float*)d_in[22];
  const float* bo         = (const float*)d_in[23];
  const float* ipa_g      = (const float*)d_in[24];
  const float* ipa_b      = (const float*)d_in[25];
  const float* tw1        = (const float*)d_in[26];
  const float* tb1        = (const float*)d_in[27];
  const float* tw2        = (const float*)d_in[28];
  const float* tb2        = (const float*)d_in[29];
  const float* tw3        = (const float*)d_in[30];
  const float* tb3        = (const float*)d_in[31];
  const float* trg        = (const float*)d_in[32];
  const float* trb        = (const float*)d_in[33];
  const float* wbbp       = (const float*)d_in[34];
  const float* bbbp       = (const float*)d_in[35];

  float* ws = (float*)d_ws;
  float* S    = ws;                       // 512*384
  float* T1   = S    + kN * kCS;
  float* T2   = T1   + kN * kCS;
  float* Q    = T2   + kN * kCS;          // 512*192
  float* KV   = Q    + kN * kQD;          // 512*384
  float* QP   = KV   + kN * kKVD;         // 512*144
  float* KVP  = QP   + kN * kQPD;         // 512*432
  float* QPG  = KVP  + kN * kKPD;         // 512*144
  float* KVPG = QPG  + kN * kQPD;         // 512*432
  float* SQQ  = KVPG + kN * kKPD;         // 512*12
  float* SQK  = SQQ  + kN * kH;           // 512*12
  float* RB   = SQK  + kN * kH;           // 512*9
  float* TB   = RB   + kN * 9;            // 512*3
  float* UPD  = TB   + kN * 3;            // 512*6
  float* OPT  = UPD  + kN * 6;            // 512*288
  float* MU   = OPT  + kN * kH * 24;      // 262144
  float* RSd  = MU   + kNN;               // 262144
  float* BM   = RSd  + kNN;               // 262144*12
  float* CAT  = BM   + (size_t)kNN * kH;  // 512*2112
  float* AB   = CAT  + (size_t)kN * kCAT; // 12*262144

  auto gemm = [&](const float* A, int lda, const float* W, int ldw,
                  const float* bias, const float* resid,
                  float* Cp, int ldc, int M, int K, int Nc, int relu) {
    const int nt = (Nc + 15) / 16;
    dim3 grid((nt + 3) / 4, M / 16);
    gemm_wmma<<<grid, dim3(128), 0, stream>>>(A, lda, W, ldw, bias, resid,
                                              Cp, ldc, K, Nc, relu);
  };

  // ---- Preamble -----------------------------------------------------------
  pair_stats<<<dim3(kNN / 8), dim3(256), 0, stream>>>(pair_rep, MU, RSd);
  ln_rows<<<dim3(kN), dim3(256), 0, stream>>>(single_rep, T1, sr_ln_g, sr_ln_b, kCS);
  gemm(T1, kCS, sr_w, kCS, sr_b, nullptr, S, kCS, kN, kCS, kCS, 0);
  copy_f32<<<dim3((kN * 9 + 255) / 256), dim3(256), 0, stream>>>(RB, rot, kN * 9);
  copy_f32<<<dim3((kN * 3 + 255) / 256), dim3(256), 0, stream>>>(TB, trans, kN * 3);

  // ---- Layers -------------------------------------------------------------
  for (int l = 0; l < kL; ++l) {
    gemm(S, kCS, wq   + (size_t)l * kCS * kQD,  kQD,  bq   + l * kQD,  nullptr, Q,   kQD,  kN, kCS, kQD,  0);
    gemm(S, kCS, wkv  + (size_t)l * kCS * kKVD, kKVD, bkv  + l * kKVD, nullptr, KV,  kKVD, kN, kCS, kKVD, 0);
    gemm(S, kCS, wqp  + (size_t)l * kCS * kQPD, kQPD, bqp  + l * kQPD, nullptr, QP,  kQPD, kN, kCS, kQPD, 0);
    gemm(S, kCS, wkvp + (size_t)l * kCS * kKPD, kKPD, bkvp + l * kKPD, nullptr, KVP, kKPD, kN, kCS, kKPD, 0);

    point_transform<<<dim3(kN), dim3(160), 0, stream>>>(QP, KVP, RB, TB, QPG, KVPG);
    point_sums<<<dim3((kN * kH + 127) / 128), dim3(128), 0, stream>>>(QPG, KVPG, SQQ, SQK);

    zb_gemm<<<dim3(kNN / 64), dim3(128), 0, stream>>>(
        pair_rep, MU, RSd, pr_ln_g, pr_ln_b,
        wb + (size_t)l * kCZ * kH, bpb + l * kH, BM);

    attn_logits<<<dim3(kN / 16, kN / 16, kH), dim3(32), 0, stream>>>(
        Q, KV, QPG, KVPG, SQQ, SQK, BM, maskp, head_w + l * kH, AB);

    softmax_rows<<<dim3(kH * kN), dim3(256), 0, stream>>>(AB);

    av_gemm<<<dim3(kN / 16, kH), dim3(96), 0, stream>>>(AB, KV, KVPG, CAT, OPT);
    opair_gemm<<<dim3(kN), dim3(256), 0, stream>>>(AB, pair_rep, MU, RSd,
                                                   pr_ln_g, pr_ln_b, CAT);
    opt_post<<<dim3((kN * kH * kPV + 127) / 128), dim3(128), 0, stream>>>(OPT, RB, TB, CAT);

    // s = s + cat @ wo + bo ; LN
    gemm(CAT, kCAT, wo + (size_t)l * kCAT * kCS, kCS, bo + l * kCS, S, S, kCS, kN, kCAT, kCS, 0);
    ln_rows<<<dim3(kN), dim3(256), 0, stream>>>(S, S, ipa_g + l * kCS, ipa_b + l * kCS, kCS);

    // transition
    gemm(S,  kCS, tw1 + (size_t)l * kCS * kCS, kCS, tb1 + l * kCS, nullptr, T1, kCS, kN, kCS, kCS, 1);
    gemm(T1, kCS, tw2 + (size_t)l * kCS * kCS, kCS, tb2 + l * kCS, nullptr, T2, kCS, kN, kCS, kCS, 1);
    gemm(T2, kCS, tw3 + (size_t)l * kCS * kCS, kCS, tb3 + l * kCS, S,       S,  kCS, kN, kCS, kCS, 0);
    ln_rows<<<dim3(kN), dim3(256), 0, stream>>>(S, S, trg + l * kCS, trb + l * kCS, kCS);

    // backbone update
    gemm(S, kCS, wbbp + (size_t)l * kCS * 6, 6, bbbp + l * 6, nullptr, UPD, 6, kN, kCS, 6, 0);
    frame_update<<<dim3(2), dim3(256), 0, stream>>>(RB, TB, UPD);
  }

  // ---- Output: (R, t) concatenated ---------------------------------------
  float* outp = (float*)d_out;
  copy_f32<<<dim3((kN * 9 + 255) / 256), dim3(256), 0, stream>>>(outp, RB, kN * 9);
  copy_f32<<<dim3((kN * 3 + 255) / 256), dim3(256), 0, stream>>>(outp + kN * 9, TB, kN * 3);
}